// Network_46368466927913
// MI455X (gfx1250) — compile-verified
//
#include <hip/hip_runtime.h>
#include <math.h>

typedef float v2f __attribute__((ext_vector_type(2)));
typedef float v8f __attribute__((ext_vector_type(8)));

// ---------------------------------------------------------------------------
// Fused 1x1-conv GEMM via fp32 WMMA:  Y[M,Co] = act((X[M,Ci] @ W[Co,Ci]^T)*s + b)
// mode: 0 = raw, 1 = scale/bias + leaky(0.2), 2 = scale/bias (no act), 3 = +bias
// Requirements (hold everywhere in this net): M % 16 == 0, Ci even, Ci >= 8.
// One wave -> one 16 x (16*NT) strip of Y; K stepped by 4 (V_WMMA_F32_16X16X4_F32).
// A loaded once per K-step, reused across NT column tiles. Register double-
// buffering: iteration n+1's fragments are loaded before iteration n's WMMAs so
// loads stay in flight behind the matrix ops (partial s_wait_loadcnt).
// ---------------------------------------------------------------------------
template <int NT>
__global__ __launch_bounds__(256) void k_gemm_wmma(
    const float* __restrict__ X, const float* __restrict__ W,
    const float* __restrict__ scale, const float* __restrict__ bias,
    float* __restrict__ Y, int M, int Ci, int Co, int mode)
{
    const int tilesM = M >> 4;
    const int tilesN = (Co + 16 * NT - 1) / (16 * NT);
    const int nTiles = tilesM * tilesN;
    const int tile = blockIdx.x * (blockDim.x >> 5) + (threadIdx.x >> 5);
    if (tile >= nTiles) return;                    // wave-uniform exit

    const int tm   = tile % tilesM;
    const int tn   = tile / tilesM;
    const int lane = threadIdx.x & 31;
    const int half = lane >> 4;                    // 0: lanes 0-15, 1: lanes 16-31
    const int l15  = lane & 15;
    const int row  = (tm << 4) + l15;              // A row (always < M)
    const int kh   = half << 1;                    // K offset inside a 4-step

    const float* __restrict__ xr = X + (size_t)row * Ci + kh;
    const float* wr[NT];
    int col[NT];
#pragma unroll
    for (int t = 0; t < NT; ++t) {
        int c  = (tn * NT + t) * 16 + l15;         // output channel
        col[t] = c;
        int cc = (c < Co) ? c : (Co - 1);          // clamp: loads valid, unstored
        wr[t]  = W + (size_t)cc * Ci + kh;
    }

    v8f acc[NT];
#pragma unroll
    for (int t = 0; t < NT; ++t) acc[t] = (v8f){0.f,0.f,0.f,0.f,0.f,0.f,0.f,0.f};

    const int CiMain = Ci & ~3;                    // >= 8 for every GEMM here

    // -- software pipeline: prologue loads ----------------------------------
    v2f a0 = *(const v2f*)(xr);
    v2f b0[NT];
#pragma unroll
    for (int t = 0; t < NT; ++t) b0[t] = *(const v2f*)(wr[t]);

    for (int k0 = 4; k0 < CiMain; k0 += 4) {
        v2f a1 = *(const v2f*)(xr + k0);           // next step's fragments
        v2f b1[NT];
#pragma unroll
        for (int t = 0; t < NT; ++t) b1[t] = *(const v2f*)(wr[t] + k0);
#pragma unroll
        for (int t = 0; t < NT; ++t)               // consume resident fragments
            acc[t] = __builtin_amdgcn_wmma_f32_16x16x4_f32(
                false, a0, false, b0[t], (short)0, acc[t], false, false);
        a0 = a1;
#pragma unroll
        for (int t = 0; t < NT; ++t) b0[t] = b1[t];
    }
#pragma unroll
    for (int t = 0; t < NT; ++t)                   // drain
        acc[t] = __builtin_amdgcn_wmma_f32_16x16x4_f32(
            false, a0, false, b0[t], (short)0, acc[t], false, false);

    if (CiMain < Ci) {                             // Ci % 4 == 2 tail (Ci==10)
        const int  k  = CiMain + kh;
        const bool ok = (k < Ci);
        const int  kc = ok ? k : (Ci - 2);         // clamped, valid, 8B aligned
        v2f a = *(const v2f*)(X + (size_t)row * Ci + kc);
        if (!ok) { a.x = 0.f; a.y = 0.f; }         // cndmask, no EXEC change
#pragma unroll
        for (int t = 0; t < NT; ++t) {
            v2f b = *(const v2f*)(wr[t] - kh + kc);
            acc[t] = __builtin_amdgcn_wmma_f32_16x16x4_f32(
                false, a, false, b, (short)0, acc[t], false, false);
        }
    }

    const int mbase = (tm << 4) + (half << 3);     // D: vgpr j -> M = mbase + j
#pragma unroll
    for (int t = 0; t < NT; ++t) {
        const int c = col[t];
        if (c < Co) {
            float s = 1.f, bb = 0.f;
            if (mode == 1 || mode == 2) { s = scale[c]; bb = bias[c]; }
            else if (mode == 3)         { bb = bias[c]; }
#pragma unroll
            for (int j = 0; j < 8; ++j) {
                float v = acc[t][j] * s + bb;
                if (mode == 1) v = (v > 0.f) ? v : 0.2f * v;
                Y[(size_t)(mbase + j) * Co + c] = v;
            }
        }
    }
}

// ---------------------------------------------------------------------------
// fc0: y = leaky(W x + b, 0.01) * scale + bias   (6 -> 8, one thread per point)
// ---------------------------------------------------------------------------
__global__ void k_fc0(const float* __restrict__ f, const float* __restrict__ W,
                      const float* __restrict__ b, const float* __restrict__ scale,
                      const float* __restrict__ bias, float* __restrict__ out, int P)
{
    int p = blockIdx.x * blockDim.x + threadIdx.x;
    if (p >= P) return;
    float in[6];
#pragma unroll
    for (int c = 0; c < 6; ++c) in[c] = f[(size_t)p * 6 + c];
#pragma unroll
    for (int o = 0; o < 8; ++o) {
        float t = b[o];
#pragma unroll
        for (int c = 0; c < 6; ++c) t += in[c] * W[o * 6 + c];
        t = (t > 0.f) ? t : 0.01f * t;
        out[(size_t)p * 8 + o] = t * scale[o] + bias[o];
    }
}

// ---------------------------------------------------------------------------
// Relative-position encoding: out[b,n,k,10] = {dist, rel(3), center(3), nbr(3)}
// ---------------------------------------------------------------------------
__global__ void k_relpos(const float* __restrict__ xyz, const int* __restrict__ neigh,
                         float* __restrict__ out, int B, int N, int K)
{
    int t = blockIdx.x * blockDim.x + threadIdx.x;
    int total = B * N * K;
    if (t >= total) return;
    int pn  = t / K;                 // b*N + n
    int b   = pn / N;
    int idx = neigh[t];
    const float* c  = xyz + (size_t)pn * 3;
    const float* nb = xyz + ((size_t)b * N + idx) * 3;
    float rx = c[0] - nb[0], ry = c[1] - nb[1], rz = c[2] - nb[2];
    float dis = sqrtf(rx * rx + ry * ry + rz * rz);
    float* o = out + (size_t)t * 10;
    o[0] = dis; o[1] = rx;   o[2] = ry;   o[3] = rz;
    o[4] = c[0]; o[5] = c[1]; o[6] = c[2];
    o[7] = nb[0]; o[8] = nb[1]; o[9] = nb[2];
}

// ---------------------------------------------------------------------------
// out[b,n,k, 0:C1] = src[b, neigh[b,n,k], :] ; out[b,n,k, C1:C1+C2] = extra[b,n,k,:]
// ---------------------------------------------------------------------------
__global__ void k_gather_concat(const float* __restrict__ src, const int* __restrict__ neigh,
                                const float* __restrict__ extra, float* __restrict__ out,
                                int B, int Nsrc, int N, int K, int C1, int C2, int total)
{
    int t = blockIdx.x * blockDim.x + threadIdx.x;
    if (t >= total) return;
    int C = C1 + C2;
    int c = t % C;
    int p = t / C;                   // b*N*K + n*K + k
    int b = p / (N * K);
    int idx = neigh[p];
    out[t] = (c < C1) ? src[((size_t)b * Nsrc + idx) * C1 + c]
                      : extra[(size_t)p * C2 + (c - C1)];
}

// ---------------------------------------------------------------------------
// Attentive pooling reduce: out[p,c] = sum_k f[p,k,c] * softmax_k(att[p,k,c]); K=16
// ---------------------------------------------------------------------------
__global__ void k_attpool(const float* __restrict__ f, const float* __restrict__ att,
                          float* __restrict__ out, int P, int d)
{
    int t = blockIdx.x * blockDim.x + threadIdx.x;
    if (t >= P * d) return;
    int c = t % d;
    int p = t / d;
    const float* ap = att + (size_t)p * 16 * d + c;
    const float* fp = f   + (size_t)p * 16 * d + c;
    float av[16];
    float mx = -3.4e38f;
#pragma unroll
    for (int k = 0; k < 16; ++k) { av[k] = ap[(size_t)k * d]; mx = fmaxf(mx, av[k]); }
    float se = 0.f, ws = 0.f;
#pragma unroll
    for (int k = 0; k < 16; ++k) {
        float e = __expf(av[k] - mx);
        se += e;
        ws += e * fp[(size_t)k * d];
    }
    out[t] = ws / se;
}

// ---------------------------------------------------------------------------
// out = leaky(a + b, 0.2)
// ---------------------------------------------------------------------------
__global__ void k_add_leaky(const float* __restrict__ a, const float* __restrict__ b,
                            float* __restrict__ o, int n)
{
    int t = blockIdx.x * blockDim.x + threadIdx.x;
    if (t >= n) return;
    float v = a[t] + b[t];
    o[t] = (v > 0.f) ? v : 0.2f * v;
}

// ---------------------------------------------------------------------------
// Random-sample pooling: out[b,m,c] = max_k f[b, sub[b,m,k], c]; K=16
// ---------------------------------------------------------------------------
__global__ void k_sub_max(const float* __restrict__ f, const int* __restrict__ sub,
                          float* __restrict__ out, int B, int Nsrc, int M, int C, int total)
{
    int t = blockIdx.x * blockDim.x + threadIdx.x;
    if (t >= total) return;
    int c = t % C;
    int p = t / C;                   // b*M + m
    int b = p / M;
    const int* si = sub + (size_t)p * 16;
    float mx = -3.4e38f;
#pragma unroll
    for (int k = 0; k < 16; ++k)
        mx = fmaxf(mx, f[((size_t)b * Nsrc + si[k]) * C + c]);
    out[t] = mx;
}

// ---------------------------------------------------------------------------
// three_interpolate (uniform 1/3): h[b,na, off+c] = mean_j f[b, itp[b,na,j], c]
// ---------------------------------------------------------------------------
__global__ void k_interp(const float* __restrict__ f, const int* __restrict__ itp,
                         float* __restrict__ h, int B, int Nsrc, int NA, int C,
                         int Ctot, int off, int total)
{
    int t = blockIdx.x * blockDim.x + threadIdx.x;
    if (t >= total) return;
    int c = t % C;
    int p = t / C;                   // b*NA + na
    int b = p / NA;
    const int* ii = itp + (size_t)p * 3;
    float s = f[((size_t)b * Nsrc + ii[0]) * C + c]
            + f[((size_t)b * Nsrc + ii[1]) * C + c]
            + f[((size_t)b * Nsrc + ii[2]) * C + c];
    h[(size_t)p * Ctot + off + c] = s * (1.f / 3.f);
}

// ===========================================================================
extern "C" void kernel_launch(void* const* d_in, const int* in_sizes, int n_in,
                              void* d_out, int out_size, void* d_ws, size_t ws_size,
                              hipStream_t stream)
{
    (void)in_sizes; (void)n_in; (void)out_size; (void)ws_size;
    static const int DOUT[5] = {16, 64, 128, 256, 512};
    static const int NS[6]   = {16384, 4096, 1024, 256, 64, 32};
    const int Bc = 2, K = 16, NA = 2048;

    // ---- inputs -----------------------------------------------------------
    const float* features = (const float*)d_in[0];
    const float* xyz[5]; const int *neigh[5], *sub[5], *interp[5];
    for (int i = 0; i < 5; ++i) {
        xyz[i]    = (const float*)d_in[1 + 4 * i];
        neigh[i]  = (const int*)  d_in[2 + 4 * i];
        sub[i]    = (const int*)  d_in[3 + 4 * i];
        interp[i] = (const int*)  d_in[4 + 4 * i];
    }
    // ---- params (JAX tree-leaves order: sorted nested dict keys) ----------
    auto P = [&](int i) { return (const float*)d_in[i]; };
    const int PB = 21;                               // dec[0..2]{W,bias,scale}
    auto encB = [&](int i) { return PB + 9 + 23 * i; };
    // enc block offsets: 0 ap1_fc.W | 1..3 ap1_mlp{W,b,s} | 4 ap2_fc.W
    // 5..7 ap2_mlp{W,b,s} | 8..10 bb_mlp1 | 11..13 bb_mlp2 | 14..16 mlp1
    // 17..19 mlp2 | 20..22 shortcut
    const int FB = PB + 9 + 115;                     // fc0_W, fc0_b, fc0_bias, fc0_scale
    const float *fc0W = P(FB), *fc0b = P(FB + 1), *fc0bias = P(FB + 2), *fc0scale = P(FB + 3);
    const float *fc1W = P(FB + 4), *fc1b = P(FB + 5), *fc1s = P(FB + 6);
    const float *fc2W = P(FB + 7), *fc2b = P(FB + 8), *fc2s = P(FB + 9);
    const float *fc3W = P(FB + 10), *fc3b = P(FB + 11);

    // ---- workspace (bump allocator, 64B-aligned slices) -------------------
    float* w = (float*)d_ws;
    size_t off = 0;
    auto alloc = [&](size_t n) { float* p = w + off; off += (n + 15) & ~(size_t)15; return p; };
    float* FEATA = alloc(262144);   float* FEATB = alloc(262144);
    float* FPC   = alloc(262144);   float* NB10  = alloc(5242880);
    float* FXYZ  = alloc(4194304);  float* FXYZ2 = alloc(4194304);
    float* FCAT  = alloc(8388608);  float* ATT   = alloc(8388608);
    float* POOL  = alloc(524288);   float* AGG1  = alloc(262144);
    float* AGG2  = alloc(524288);
    float* MAINB = alloc(1048576);  float* SCB   = alloc(1048576);
    float* FENC  = alloc(1048576);  float* H     = alloc(7995392);
    float* DECA  = alloc(1048576);  float* DECB  = alloc(1048576);

    auto blocks = [](long long n) { return (unsigned)((n + 255) / 256); };
    auto gemm = [&](const float* X, const float* Wp, const float* sc, const float* bi,
                    float* Y, int M, int Ci, int Co, int mode) {
        if (Co % 64 == 0) {
            int tiles = (M >> 4) * (Co >> 6);
            k_gemm_wmma<4><<<(tiles + 7) / 8, 256, 0, stream>>>(X, Wp, sc, bi, Y, M, Ci, Co, mode);
        } else if (Co % 32 == 0) {
            int tiles = (M >> 4) * (Co >> 5);
            k_gemm_wmma<2><<<(tiles + 7) / 8, 256, 0, stream>>>(X, Wp, sc, bi, Y, M, Ci, Co, mode);
        } else {
            int tiles = (M >> 4) * ((Co + 15) >> 4);
            k_gemm_wmma<1><<<(tiles + 7) / 8, 256, 0, stream>>>(X, Wp, sc, bi, Y, M, Ci, Co, mode);
        }
    };

    // ---- fc0 --------------------------------------------------------------
    const int P0 = Bc * NS[0];
    k_fc0<<<blocks(P0), 256, 0, stream>>>(features, fc0W, fc0b, fc0scale, fc0bias, FEATA, P0);

    // ---- encoder ----------------------------------------------------------
    const float* feat = FEATA;
    float* featNext = FEATB;
    int Cin = 8, hoff = 0;
    for (int i = 0; i < 5; ++i) {
        const int N = NS[i], M = Bc * N, MK = M * K;
        const int d = DOUT[i], d2 = d / 2, d4 = 2 * d;
        const int eb = encB(i);

        gemm(feat, P(eb + 14), P(eb + 16), P(eb + 15), FPC, M, Cin, d2, 1);     // mlp1
        k_relpos<<<blocks(MK), 256, 0, stream>>>(xyz[i], neigh[i], NB10, Bc, N, K);
        gemm(NB10, P(eb + 8), P(eb + 10), P(eb + 9), FXYZ, MK, 10, d2, 1);      // bb_mlp1
        k_gather_concat<<<blocks((long long)MK * d), 256, 0, stream>>>(
            FPC, neigh[i], FXYZ, FCAT, Bc, N, N, K, d2, d2, MK * d);
        gemm(FCAT, P(eb + 0), nullptr, nullptr, ATT, MK, d, d, 0);              // ap1_fc
        k_attpool<<<blocks((long long)M * d), 256, 0, stream>>>(FCAT, ATT, POOL, M, d);
        gemm(POOL, P(eb + 1), P(eb + 3), P(eb + 2), AGG1, M, d, d2, 1);         // ap1_mlp
        gemm(FXYZ, P(eb + 11), P(eb + 13), P(eb + 12), FXYZ2, MK, d2, d2, 1);   // bb_mlp2
        k_gather_concat<<<blocks((long long)MK * d), 256, 0, stream>>>(
            AGG1, neigh[i], FXYZ2, FCAT, Bc, N, N, K, d2, d2, MK * d);
        gemm(FCAT, P(eb + 4), nullptr, nullptr, ATT, MK, d, d, 0);              // ap2_fc
        k_attpool<<<blocks((long long)M * d), 256, 0, stream>>>(FCAT, ATT, POOL, M, d);
        gemm(POOL, P(eb + 5), P(eb + 7), P(eb + 6), AGG2, M, d, d, 1);          // ap2_mlp
        gemm(AGG2, P(eb + 17), P(eb + 19), P(eb + 18), MAINB, M, d, d4, 2);     // mlp2
        gemm(feat, P(eb + 20), P(eb + 22), P(eb + 21), SCB, M, Cin, d4, 2);     // shortcut
        k_add_leaky<<<blocks((long long)M * d4), 256, 0, stream>>>(MAINB, SCB, FENC, M * d4);

        const int Mn = Bc * NS[i + 1];
        k_sub_max<<<blocks((long long)Mn * d4), 256, 0, stream>>>(
            FENC, sub[i], featNext, Bc, N, NS[i + 1], d4, Mn * d4);
        k_interp<<<blocks((long long)Bc * NA * d4), 256, 0, stream>>>(
            FENC, interp[i], H, Bc, N, NA, d4, 1952, hoff, Bc * NA * d4);
        hoff += d4;

        feat = featNext;
        featNext = (feat == FEATA) ? FEATB : FEATA;
        Cin = d4;
    }

    // ---- decoder + head ---------------------------------------------------
    const int MD = Bc * NA;
    gemm(H,    P(PB + 0), P(PB + 2), P(PB + 1), DECA, MD, 1952, 256, 1);
    gemm(DECA, P(PB + 3), P(PB + 5), P(PB + 4), DECB, MD, 256, 128, 1);
    gemm(DECB, P(PB + 6), P(PB + 8), P(PB + 7), DECA, MD, 128, 64, 1);
    gemm(DECA, fc1W, fc1s, fc1b, DECB, MD, 64, 64, 1);
    gemm(DECB, fc2W, fc2s, fc2b, DECA, MD, 64, 32, 1);
    gemm(DECA, fc3W, nullptr, fc3b, (float*)d_out, MD, 32, 13, 3);              // logits
}